// MDTA_FOR_VIDEO_49237505081691
// MI455X (gfx1250) — compile-verified
//
#include <hip/hip_runtime.h>
#include <hip/hip_bf16.h>
#include <math.h>

// ---------------------------------------------------------------------------
// MDTA-for-video pipeline on gfx1250 (MI455X).
// Heavy convs: implicit-GEMM with V_WMMA_F32_16X16X4_F32 (fp32 matrix pipe).
// 64x64 block tile, 8 waves, 2 subtiles/wave (shared A fragment).
// ---------------------------------------------------------------------------

typedef float v2f __attribute__((ext_vector_type(2)));
typedef float v8f __attribute__((ext_vector_type(8)));

#define B_  4
#define C_  192
#define C2_ 384
#define H_  96
#define W_  96
#define HW_ 9216
#define NH_ 8
#define CH_ 24     // C / NH
#define G_  8
#define CG_ 24     // C / G
#define PH_ 48
#define QH_ 46
#define QHW_ (46*46)

__device__ __forceinline__ float sigmoidf_(float v) { return 1.0f / (1.0f + __expf(-v)); }

// ---------------------------------------------------------------------------
// Conv (kxk, stride1, pad p) as implicit GEMM with WMMA f32 16x16x4.
// M = Cout, N = Hout*Wout (per batch), K = Cin*k*k (always a multiple of 16
// for this model: 192 / 3456).
// Block: 256 threads = 8 wave32s. Block tile 64(M) x 64(N); each wave owns
// one 16-row M strip and two 16-col N subtiles -> A fragment reused across
// 2 WMMAs. A/B staged via LDS in K-steps of 16 (8 WMMA per step per wave).
// ---------------------------------------------------------------------------
#define TK 16
#define TM 64
#define TN 64

__global__ __launch_bounds__(256)
void conv_gemm_wmma(const float* __restrict__ in, const float* __restrict__ wgt,
                    const float* __restrict__ bias, float* __restrict__ out,
                    int Cin, int Cout, int Hin, int Win, int Hout, int Wout,
                    int ksize, int pad, int relu_in)
{
    const int Npos = Hout * Wout;
    const int ksz2 = ksize * ksize;
    const int K    = Cin * ksz2;
    const int b    = blockIdx.z;

    __shared__ float lA[TK][TM];
    __shared__ float lB[TK][TN];

    const int tid  = threadIdx.x;
    const int lane = tid & 31;
    const int wid  = tid >> 5;               // 0..7
    const int mb   = (wid & 3) * 16;         // wave's M sub-strip in block tile
    const int nb0  = (wid >> 2) * 32;        // wave's first N subtile
    const int nb1  = nb0 + 16;               // wave's second N subtile

    const int m0 = blockIdx.y * TM;
    const int n0 = blockIdx.x * TN;

    const float* inB = in + (size_t)b * Cin * Hin * Win;

    const int l16 = lane & 15;
    const int hi2 = (lane >> 4) << 1;        // 0: K+0/K+1  |  2: K+2/K+3

    v8f acc0 = {0.f,0.f,0.f,0.f,0.f,0.f,0.f,0.f};
    v8f acc1 = {0.f,0.f,0.f,0.f,0.f,0.f,0.f,0.f};

    for (int k0 = 0; k0 < K; k0 += TK) {
        // ---- stage A (weights, [K x M] in LDS): 1024 elems / 256 threads ----
        #pragma unroll
        for (int i = 0; i < 4; ++i) {
            int e  = tid + i * 256;
            int kl = e >> 6;                 // 0..15
            int m  = e & 63;
            int kg = k0 + kl;
            int mg = m0 + m;
            int mgc = min(mg, Cout - 1);
            float v = wgt[(size_t)mgc * K + kg];      // unconditional (clamped) load
            lA[kl][m] = (mg < Cout) ? v : 0.f;
        }
        // ---- stage B (im2col input, [K x N] in LDS) ----
        #pragma unroll
        for (int i = 0; i < 4; ++i) {
            int e  = tid + i * 256;
            int kl = e >> 6;
            int n  = e & 63;
            int kg = k0 + kl;
            int ng = n0 + n;
            int ngc = min(ng, Npos - 1);
            int ci = kg / ksz2;
            int r  = kg - ci * ksz2;
            int ky = r / ksize;
            int kx = r - ky * ksize;
            int oy = ngc / Wout;
            int ox = ngc - oy * Wout;
            int iy = oy + ky - pad;
            int ix = ox + kx - pad;
            bool ok = (ng < Npos) & (iy >= 0) & (iy < Hin) & (ix >= 0) & (ix < Win);
            int cy = min(max(iy, 0), Hin - 1);
            int cx = min(max(ix, 0), Win - 1);
            float v = inB[((size_t)ci * Hin + cy) * Win + cx];   // unconditional load
            if (relu_in) v = fmaxf(v, 0.f);
            lB[kl][n] = ok ? v : 0.f;
        }
        // prefetch next K-tile while this one is consumed
        if (k0 + TK < K) {
            int mg = min(m0 + (tid & 63), Cout - 1);
            __builtin_prefetch(&wgt[(size_t)mg * K + k0 + TK], 0, 1);
            int cin = min((k0 + TK) / ksz2, Cin - 1);
            __builtin_prefetch(&inB[(size_t)cin * Hin * Win + (tid & 63)], 0, 1);
        }
        __syncthreads();

        // ---- 8 x WMMA (K=4 each): A fragment reused across 2 N subtiles ----
        #pragma unroll
        for (int j = 0; j < TK; j += 4) {
            v2f a, b0, b1;
            a.x  = lA[j + hi2 + 0][mb + l16];
            a.y  = lA[j + hi2 + 1][mb + l16];
            b0.x = lB[j + hi2 + 0][nb0 + l16];
            b0.y = lB[j + hi2 + 1][nb0 + l16];
            b1.x = lB[j + hi2 + 0][nb1 + l16];
            b1.y = lB[j + hi2 + 1][nb1 + l16];
            acc0 = __builtin_amdgcn_wmma_f32_16x16x4_f32(
                       false, a, false, b0, (short)0, acc0, false, false);
            acc1 = __builtin_amdgcn_wmma_f32_16x16x4_f32(
                       false, a, false, b1, (short)0, acc1, false, false);
        }
        __syncthreads();
    }

    // ---- epilogue: C/D layout: VGPR r -> M=r (lanes 0-15) / M=r+8 (lanes 16-31)
    const int rhalf = (lane >> 4) * 8;
    #pragma unroll
    for (int r = 0; r < 8; ++r) {
        int mg = m0 + mb + r + rhalf;
        if (mg < Cout) {
            float bv = bias ? bias[mg] : 0.f;
            int ng0 = n0 + nb0 + l16;
            int ng1 = n0 + nb1 + l16;
            if (ng0 < Npos)
                out[((size_t)b * Cout + mg) * Npos + ng0] = acc0[r] + bv;
            if (ng1 < Npos)
                out[((size_t)b * Cout + mg) * Npos + ng1] = acc1[r] + bv;
        }
    }
}

// ---------------------------------------------------------------------------
// Grouped depthwise-ish conv: groups=192, each group 2-in -> 2-out, 3x3 pad 1.
// Splits output into q (ch 0..191) and v (ch 192..383).
// ---------------------------------------------------------------------------
__global__ void dwconv_qkv(const float* __restrict__ qkv, const float* __restrict__ w,
                           float* __restrict__ q, float* __restrict__ v)
{
    size_t idx = (size_t)blockIdx.x * blockDim.x + threadIdx.x;
    const size_t total = (size_t)B_ * C2_ * HW_;
    if (idx >= total) return;
    int n  = (int)(idx % HW_);
    int oc = (int)((idx / HW_) % C2_);
    int b  = (int)(idx / ((size_t)HW_ * C2_));
    int y  = n / W_, x = n % W_;
    int g  = oc >> 1;
    float acc = 0.f;
    #pragma unroll
    for (int i = 0; i < 2; ++i) {
        const float* src = qkv + ((size_t)b * C2_ + (g * 2 + i)) * HW_;
        const float* wk  = w + ((size_t)oc * 2 + i) * 9;
        #pragma unroll
        for (int t = 0; t < 9; ++t) {
            int ky = t / 3, kx = t % 3;
            int iy = y + ky - 1, ix = x + kx - 1;
            if (iy >= 0 && iy < H_ && ix >= 0 && ix < W_)
                acc += src[iy * W_ + ix] * wk[t];
        }
    }
    if (oc < C_) q[((size_t)b * C_ + oc) * HW_ + n] = acc;
    else         v[((size_t)b * C_ + (oc - C_)) * HW_ + n] = acc;
}

// cat = concat([q, y], axis=1)
__global__ void concat_qy(const float* __restrict__ q, const float* __restrict__ y,
                          float* __restrict__ cat)
{
    size_t idx = (size_t)blockIdx.x * blockDim.x + threadIdx.x;
    const size_t total = (size_t)B_ * C2_ * HW_;
    if (idx >= total) return;
    int n = (int)(idx % HW_);
    int c = (int)((idx / HW_) % C2_);
    int b = (int)(idx / ((size_t)HW_ * C2_));
    float v = (c < C_) ? q[((size_t)b * C_ + c) * HW_ + n]
                       : y[((size_t)b * C_ + (c - C_)) * HW_ + n];
    cat[idx] = v;
}

// 2x2 average pool, stride 2 (96 -> 48)
__global__ void avgpool2(const float* __restrict__ cat, float* __restrict__ pooled)
{
    size_t idx = (size_t)blockIdx.x * blockDim.x + threadIdx.x;
    const size_t total = (size_t)B_ * C2_ * PH_ * PH_;
    if (idx >= total) return;
    int px = (int)(idx % PH_);
    int py = (int)((idx / PH_) % PH_);
    int c  = (int)((idx / (PH_ * PH_)) % C2_);
    int b  = (int)(idx / ((size_t)PH_ * PH_ * C2_));
    const float* s = cat + ((size_t)b * C2_ + c) * HW_;
    int y = py * 2, x = px * 2;
    pooled[idx] = 0.25f * (s[y * W_ + x] + s[y * W_ + x + 1] +
                           s[(y + 1) * W_ + x] + s[(y + 1) * W_ + x + 1]);
}

// sig = sigmoid(cat + nearest_resize(k2out, 96,96))
__global__ void sig_gate(const float* __restrict__ cat, const float* __restrict__ k2o,
                         float* __restrict__ sig)
{
    size_t idx = (size_t)blockIdx.x * blockDim.x + threadIdx.x;
    const size_t total = (size_t)B_ * C2_ * HW_;
    if (idx >= total) return;
    int n = (int)(idx % HW_);
    int c = (int)((idx / HW_) % C2_);
    int b = (int)(idx / ((size_t)HW_ * C2_));
    int y = n / W_, x = n % W_;
    int iy = (y * QH_) / H_;
    int ix = (x * QH_) / W_;
    float v = cat[idx] + k2o[((size_t)b * C2_ + c) * QHW_ + iy * QH_ + ix];
    sig[idx] = sigmoidf_(v);
}

// a *= b (elementwise, in place)
__global__ void elem_mul(float* __restrict__ a, const float* __restrict__ b, size_t total)
{
    size_t idx = (size_t)blockIdx.x * blockDim.x + threadIdx.x;
    if (idx >= total) return;
    a[idx] = a[idx] * b[idx];
}

// ---------------------------------------------------------------------------
// Deformable conv (3x3, G=8 groups, 24 ch/group). One block per (b,g,pos-chunk);
// group weights (24x24x9 = 20.7 KB) staged in LDS.
// ---------------------------------------------------------------------------
__global__ __launch_bounds__(256)
void deform_kernel(const float* __restrict__ q, const float* __restrict__ offset,
                   const float* __restrict__ dw, const float* __restrict__ db,
                   float* __restrict__ feat)
{
    const int g = blockIdx.y;
    const int b = blockIdx.z;
    const int tid = threadIdx.x;

    __shared__ float wsh[CG_ * CG_ * 9];   // [co_local][ci][tap]
    for (int i = tid; i < CG_ * CG_ * 9; i += 256)
        wsh[i] = dw[((size_t)(g * CG_) * CG_ * 9) + i];
    __syncthreads();

    int n = blockIdx.x * 256 + tid;
    if (n >= HW_) return;
    int y = n / W_, x = n % W_;

    float acc[CG_];
    #pragma unroll
    for (int co = 0; co < CG_; ++co) acc[co] = db[g * CG_ + co];

    const float* offB = offset + (size_t)b * 18 * HW_;
    const float* qB   = q + ((size_t)b * C_ + g * CG_) * HW_;

    for (int tap = 0; tap < 9; ++tap) {
        int ky = tap / 3, kx = tap % 3;
        float dy = offB[(size_t)(tap * 2 + 0) * HW_ + n];
        float dx = offB[(size_t)(tap * 2 + 1) * HW_ + n];
        float m  = sigmoidf_(offB[(size_t)tap * HW_ + n]);   // mask = sigmoid(offset[:, :9])
        float py = dy + (float)(y - 1 + ky);
        float px = dx + (float)(x - 1 + kx);
        float y0 = floorf(py), x0 = floorf(px);
        float fy = py - y0, fx = px - x0;
        int iy0 = (int)y0, ix0 = (int)x0;
        float vy0 = (y0 >= 0.f && y0 <= (float)(H_ - 1)) ? 1.f : 0.f;
        float vy1 = (y0 + 1.f >= 0.f && y0 + 1.f <= (float)(H_ - 1)) ? 1.f : 0.f;
        float vx0 = (x0 >= 0.f && x0 <= (float)(W_ - 1)) ? 1.f : 0.f;
        float vx1 = (x0 + 1.f >= 0.f && x0 + 1.f <= (float)(W_ - 1)) ? 1.f : 0.f;
        int cy0 = min(max(iy0, 0), H_ - 1), cy1 = min(max(iy0 + 1, 0), H_ - 1);
        int cx0 = min(max(ix0, 0), W_ - 1), cx1 = min(max(ix0 + 1, 0), W_ - 1);
        float w00 = (1.f - fy) * (1.f - fx) * vy0 * vx0;
        float w01 = (1.f - fy) * fx         * vy0 * vx1;
        float w10 = fy         * (1.f - fx) * vy1 * vx0;
        float w11 = fy         * fx         * vy1 * vx1;

        for (int ci = 0; ci < CG_; ++ci) {
            const float* s = qB + (size_t)ci * HW_;
            float val = s[cy0 * W_ + cx0] * w00 + s[cy0 * W_ + cx1] * w01 +
                        s[cy1 * W_ + cx0] * w10 + s[cy1 * W_ + cx1] * w11;
            val *= m;
            const float* wr = &wsh[ci * 9 + tap];
            #pragma unroll
            for (int co = 0; co < CG_; ++co) acc[co] += val * wr[co * CG_ * 9];
        }
    }
    #pragma unroll
    for (int co = 0; co < CG_; ++co)
        feat[((size_t)b * C_ + g * CG_ + co) * HW_ + n] = acc[co];
}

// ---------------------------------------------------------------------------
// Attention: row inverse-L2-norms, logits, softmax, apply.
// ---------------------------------------------------------------------------
__global__ __launch_bounds__(256)
void rownorm(const float* __restrict__ t, float* __restrict__ invn)
{
    int row = blockIdx.x;               // B*C = 768 rows of length HW
    const float* s = t + (size_t)row * HW_;
    float ss = 0.f;
    for (int i = threadIdx.x; i < HW_; i += 256) { float v = s[i]; ss += v * v; }
    __shared__ float red[256];
    red[threadIdx.x] = ss;
    __syncthreads();
    for (int st = 128; st > 0; st >>= 1) {
        if (threadIdx.x < st) red[threadIdx.x] += red[threadIdx.x + st];
        __syncthreads();
    }
    if (threadIdx.x == 0)
        invn[row] = 1.f / fmaxf(sqrtf(red[0]), 1e-12f);
}

__global__ __launch_bounds__(256)
void attn_logits(const float* __restrict__ q, const float* __restrict__ k,
                 const float* __restrict__ invq, const float* __restrict__ invk,
                 const float* __restrict__ temp, float* __restrict__ logits)
{
    const int c = blockIdx.x;   // 0..23
    const int h = blockIdx.y;
    const int b = blockIdx.z;
    const int rowq = b * C_ + h * CH_ + c;
    const float* qs = q + (size_t)rowq * HW_;
    const float* kb = k + ((size_t)b * C_ + h * CH_) * HW_;

    float part[CH_];
    #pragma unroll
    for (int d = 0; d < CH_; ++d) part[d] = 0.f;

    for (int n = threadIdx.x; n < HW_; n += 256) {
        float qv = qs[n];
        #pragma unroll
        for (int d = 0; d < CH_; ++d) part[d] += qv * kb[(size_t)d * HW_ + n];
    }

    __shared__ float red[256];
    for (int d = 0; d < CH_; ++d) {
        red[threadIdx.x] = part[d];
        __syncthreads();
        for (int st = 128; st > 0; st >>= 1) {
            if (threadIdx.x < st) red[threadIdx.x] += red[threadIdx.x + st];
            __syncthreads();
        }
        if (threadIdx.x == 0) {
            float v = red[0] * invq[rowq] * invk[b * C_ + h * CH_ + d] * temp[h];
            logits[(((size_t)b * NH_ + h) * CH_ + c) * CH_ + d] = v;
        }
        __syncthreads();
    }
}

__global__ void softmax24(float* __restrict__ logits)
{
    int row = blockIdx.x * blockDim.x + threadIdx.x;    // B*NH*24 = 768 rows
    if (row >= B_ * NH_ * CH_) return;
    float* p = logits + (size_t)row * CH_;
    float mx = p[0];
    for (int d = 1; d < CH_; ++d) mx = fmaxf(mx, p[d]);
    float s = 0.f;
    for (int d = 0; d < CH_; ++d) { float e = __expf(p[d] - mx); p[d] = e; s += e; }
    float inv = 1.f / s;
    for (int d = 0; d < CH_; ++d) p[d] *= inv;
}

__global__ __launch_bounds__(256)
void attn_apply(const float* __restrict__ attn, const float* __restrict__ v,
                float* __restrict__ out)
{
    const int h = blockIdx.y;
    const int b = blockIdx.z;
    __shared__ float a[CH_ * CH_];
    const float* ab = attn + ((size_t)b * NH_ + h) * CH_ * CH_;
    for (int i = threadIdx.x; i < CH_ * CH_; i += 256) a[i] = ab[i];
    __syncthreads();

    int n = blockIdx.x * 256 + threadIdx.x;
    if (n >= HW_) return;
    const float* vb = v + ((size_t)b * C_ + h * CH_) * HW_;
    float o[CH_];
    #pragma unroll
    for (int c = 0; c < CH_; ++c) o[c] = 0.f;
    for (int d = 0; d < CH_; ++d) {
        float vv = vb[(size_t)d * HW_ + n];
        #pragma unroll
        for (int c = 0; c < CH_; ++c) o[c] += a[c * CH_ + d] * vv;
    }
    #pragma unroll
    for (int c = 0; c < CH_; ++c)
        out[((size_t)b * C_ + h * CH_ + c) * HW_ + n] = o[c];
}

// ---------------------------------------------------------------------------
// Host-side orchestration
// ---------------------------------------------------------------------------
extern "C" void kernel_launch(void* const* d_in, const int* in_sizes, int n_in,
                              void* d_out, int out_size, void* d_ws, size_t ws_size,
                              hipStream_t stream)
{
    const float* x        = (const float*)d_in[0];
    const float* yin      = (const float*)d_in[1];
    const float* temp     = (const float*)d_in[2];
    const float* qkv_w    = (const float*)d_in[3];
    const float* qkv_cw   = (const float*)d_in[4];
    const float* proj_w   = (const float*)d_in[5];
    const float* k2_w     = (const float*)d_in[6];
    const float* k3_w     = (const float*)d_in[7];
    const float* k4_w     = (const float*)d_in[8];
    const float* deform_w = (const float*)d_in[9];
    const float* deform_b = (const float*)d_in[10];
    const float* pw_w     = (const float*)d_in[11];
    const float* pw_b     = (const float*)d_in[12];
    float* out = (float*)d_out;
    float* ws  = (float*)d_ws;

    const size_t SZ_BC2 = (size_t)B_ * C2_ * HW_;   // 14,155,776 floats
    const size_t SZ_BC  = (size_t)B_ * C_  * HW_;   //  7,077,888 floats

    // region lifetimes: A: qkv -> sig -> offset; C: cat -> feat|k; D: pooled+k2out -> sc -> attnout
    float* bufA    = ws;                    // SZ_BC2
    float* bufQ    = bufA  + SZ_BC2;        // SZ_BC
    float* bufV    = bufQ  + SZ_BC;         // SZ_BC
    float* bufC    = bufV  + SZ_BC;         // SZ_BC2
    float* bufD    = bufC  + SZ_BC2;        // SZ_BC2
    float* bufS    = bufD  + SZ_BC2;        // small tail
    float* pooled  = bufD;                                  // B*C2*48*48
    float* k2out   = bufD + (size_t)B_ * C2_ * PH_ * PH_;   // B*C2*46*46
    float* sig     = bufA;
    float* sc      = bufD;                  // k3out, then *= sig in place
    float* offs    = bufA;                  // B*18*HW (A free after sc mul)
    float* feat    = bufC;                  // SZ_BC   (C free after k3 conv)
    float* ktens   = bufC + SZ_BC;          // SZ_BC
    float* attnout = bufD;                  // SZ_BC   (D free after k4 conv)
    float* invq    = bufS;
    float* invk    = bufS + 768;
    float* logits  = bufS + 1536;           // B*NH*24*24

    const int NT  = (HW_ + TN - 1) / TN;     // 144 position tiles (96x96)
    const int NT2 = (QHW_ + TN - 1) / TN;    // 34 position tiles (46x46)
    const size_t EW = SZ_BC2;                // elementwise total over B*C2*HW

    // 1) qkv = conv1x1(x, qkv_w): 192 -> 384
    conv_gemm_wmma<<<dim3(NT, C2_ / TM, B_), 256, 0, stream>>>(
        x, qkv_w, nullptr, bufA, C_, C2_, H_, W_, H_, W_, 1, 0, 0);

    // 2) grouped dw 3x3 (groups=192, 2->2) + split into q, v
    dwconv_qkv<<<(EW + 255) / 256, 256, 0, stream>>>(bufA, qkv_cw, bufQ, bufV);

    // 3) cat = [q, y]
    concat_qy<<<(EW + 255) / 256, 256, 0, stream>>>(bufQ, yin, bufC);

    // 4) pooled = avgpool2(cat)
    avgpool2<<<(((size_t)B_ * C2_ * PH_ * PH_) + 255) / 256, 256, 0, stream>>>(bufC, pooled);

    // 5) k2out = conv3x3_valid(pooled, k2_w): 48x48 -> 46x46
    conv_gemm_wmma<<<dim3(NT2, C2_ / TM, B_), 256, 0, stream>>>(
        pooled, k2_w, nullptr, k2out, C2_, C2_, PH_, PH_, QH_, QH_, 3, 0, 0);

    // 6) sig = sigmoid(cat + resize(k2out))
    sig_gate<<<(EW + 255) / 256, 256, 0, stream>>>(bufC, k2out, sig);

    // 7) k3out = conv3x3_same(cat, k3_w): 384 -> 384   (dominant GEMM)
    conv_gemm_wmma<<<dim3(NT, C2_ / TM, B_), 256, 0, stream>>>(
        bufC, k3_w, nullptr, sc, C2_, C2_, H_, W_, H_, W_, 3, 1, 0);

    // 8) sc = k3out * sig
    elem_mul<<<(EW + 255) / 256, 256, 0, stream>>>(sc, sig, EW);

    // 9) offset = conv3x3_same(sc, k4_w): 384 -> 18
    conv_gemm_wmma<<<dim3(NT, 1, B_), 256, 0, stream>>>(
        sc, k4_w, nullptr, offs, C2_, 18, H_, W_, H_, W_, 3, 1, 0);

    // 10) feat = deform_conv(q, offset, mask, deform_w, deform_b)
    deform_kernel<<<dim3(HW_ / 256, G_, B_), 256, 0, stream>>>(
        bufQ, offs, deform_w, deform_b, feat);

    // 11) k = conv1x1(relu(feat), pw_w) + pw_b
    conv_gemm_wmma<<<dim3(NT, C_ / TM, B_), 256, 0, stream>>>(
        feat, pw_w, pw_b, ktens, C_, C_, H_, W_, H_, W_, 1, 0, 1);

    // 12) inverse L2 norms per (b,h,c) row
    rownorm<<<B_ * C_, 256, 0, stream>>>(bufQ, invq);
    rownorm<<<B_ * C_, 256, 0, stream>>>(ktens, invk);

    // 13) logits = (qn . kn^T) * temperature
    attn_logits<<<dim3(CH_, NH_, B_), 256, 0, stream>>>(bufQ, ktens, invq, invk, temp, logits);

    // 14) softmax over last dim (24)
    softmax24<<<3, 256, 0, stream>>>(logits);

    // 15) attnout = attn @ v
    attn_apply<<<dim3(HW_ / 256, NH_, B_), 256, 0, stream>>>(logits, bufV, attnout);

    // 16) out = conv1x1(attnout, proj_w)
    conv_gemm_wmma<<<dim3(NT, C_ / TM, B_), 256, 0, stream>>>(
        attnout, proj_w, nullptr, out, C_, C_, H_, W_, H_, W_, 1, 0, 0);
}